// QuantumLLPModel_49563922596464
// MI455X (gfx1250) — compile-verified
//
#include <hip/hip_runtime.h>

typedef float v2f __attribute__((ext_vector_type(2)));
typedef float v8f __attribute__((ext_vector_type(8)));

#define N_QUBITS    16
#define NUM_CLASSES 10

// Fused butterfly step: r = v + dpp_row_xor(v) in ONE VALU instruction.
// Masks 1/2/4/8 stay within each 16-lane DPP row (our two independent halves).
#define DECL_ADD_XOR(M)                                                        \
  __device__ __forceinline__ float add_xor_##M(float v) {                      \
    float r;                                                                   \
    asm("v_add_f32_dpp %0, %1, %1 row_xmask:" #M                               \
        " row_mask:0xf bank_mask:0xf bound_ctrl:1"                             \
        : "=&v"(r)                                                             \
        : "v"(v));                                                             \
    return r;                                                                  \
  }
DECL_ADD_XOR(1) DECL_ADD_XOR(2) DECL_ADD_XOR(4) DECL_ADD_XOR(8)

// out[b,s] = softmax2_s( sum_k bit_k(s) * log2(p1[b,12+k]/p0[b,12+k]) )
// (base-2 softmax == softmax; qubits 0..11 cancel in the normalization)
__global__ __launch_bounds__(256) void qllp_wmma_kernel(
    const float* __restrict__ x, const float* __restrict__ params,
    float* __restrict__ out, int batch) {
  const int lane  = threadIdx.x & 31;
  const int hh    = lane >> 4;    // 0: K=0,1 (qubits 12,13); 1: K=2,3 (qubits 14,15)
  const int m     = lane & 15;    // sample-in-tile (A) / class column (B, D)
  const int wave  = (int)((blockIdx.x * blockDim.x + threadIdx.x) >> 5);
  const int tile0 = wave * 16;    // first sample of this wave's tile
  if (tile0 >= batch) return;     // wave-uniform: EXEC stays all-1s for WMMA

  const bool valid = (m < NUM_CLASSES);

  // ---- B matrix (constant per lane): T[k,s] = (s >> (3-k)) & 1 ----
  v2f B;
  {
    const int k0 = hh * 2;
    B.x = valid ? (float)((m >> (3 - k0)) & 1) : 0.0f;
    B.y = valid ? (float)((m >> (2 - k0)) & 1) : 0.0f;
  }

  // ---- A matrix: base-2 log-odds for this lane's 2 qubits of sample tile0+m
  const int q = 12 + hh * 2;
  const float* xr = x + (size_t)(tile0 + m) * N_QUBITS + q;
  const float x0 = xr[0], x1 = xr[1];
  const float pr0 = params[q], pr1 = params[q + 1];
  const float h0 = fmaf(x0, 1.57079632679f, 0.5f * pr0);  // (pi*x + p)/2
  const float h1 = fmaf(x1, 1.57079632679f, 0.5f * pr1);
  const float c0 = __cosf(h0), sn0 = __sinf(h0);
  const float c1 = __cosf(h1), sn1 = __sinf(h1);
  // log2(sin^2/cos^2) = 2*(log2|sin| - log2|cos|); abs is a free src modifier
  float d0 = 2.0f * (__log2f(fabsf(sn0)) - __log2f(fabsf(c0)));
  float d1 = 2.0f * (__log2f(fabsf(sn1)) - __log2f(fabsf(c1)));
  // clamp: |logit| <= 3*36 = 108 -> exp2 never overflows/underflows; binds
  // only when min(p0,p1) < ~2e-11, where affected classes carry <1e-9 mass.
  d0 = fminf(fmaxf(d0, -36.0f), 36.0f);
  d1 = fminf(fmaxf(d1, -36.0f), 36.0f);
  v2f A; A.x = d0; A.y = d1;

  // invalid class lanes: bias the accumulator so exp2(logit) == 0 for them
  const float cbias = valid ? 0.0f : -150.0f;
  v8f C;
  #pragma unroll
  for (int i = 0; i < 8; ++i) C[i] = cbias;

  // ---- logits = A(16x4) x B(4x16) + C via v_wmma_f32_16x16x4_f32 ----
  C = __builtin_amdgcn_wmma_f32_16x16x4_f32(
      /*neg_a=*/false, A, /*neg_b=*/false, B,
      /*c_mod=*/(short)0, C, /*reuse_a=*/false, /*reuse_b=*/false);

  // D layout: VGPR r, lane L -> logit[sample r + 8*hh, class L&15].
  // class 0 logit == 0 exactly, so every per-sample sum >= 1 (no 0/0).
  float e[8], s[8];
  #pragma unroll
  for (int r = 0; r < 8; ++r) {
    e[r] = __builtin_amdgcn_exp2f(C[r]);   // one v_exp_f32 each
    s[r] = e[r];
  }
  #pragma unroll
  for (int r = 0; r < 8; ++r) s[r] = add_xor_1(s[r]);
  #pragma unroll
  for (int r = 0; r < 8; ++r) s[r] = add_xor_2(s[r]);
  #pragma unroll
  for (int r = 0; r < 8; ++r) s[r] = add_xor_4(s[r]);
  #pragma unroll
  for (int r = 0; r < 8; ++r) s[r] = add_xor_8(s[r]);

  #pragma unroll
  for (int r = 0; r < 8; ++r) {
    if (valid) {
      const int samp = tile0 + r + hh * 8;
      out[(size_t)samp * NUM_CLASSES + m] = __fdividef(e[r], s[r]);
    }
  }
}

extern "C" void kernel_launch(void* const* d_in, const int* in_sizes, int n_in,
                              void* d_out, int out_size, void* d_ws, size_t ws_size,
                              hipStream_t stream) {
  const float* x      = (const float*)d_in[0];
  const float* params = (const float*)d_in[1];
  float* out          = (float*)d_out;

  const int batch   = in_sizes[0] / N_QUBITS;          // 524288
  const int waves   = (batch + 15) / 16;               // one 16-sample tile per wave
  const int threads = 256;                             // 8 waves / block
  const int blocks  = (waves * 32 + threads - 1) / threads;

  qllp_wmma_kernel<<<blocks, threads, 0, stream>>>(x, params, out, batch);
}